// ResidualVQ_80315888435486
// MI455X (gfx1250) — compile-verified
//
#include <hip/hip_runtime.h>
#include <hip/hip_bf16.h>

// Residual VQ, fp32 WMMA path for MI455X (gfx1250).
//   x:         [B,N,64]  fp32   (B*N = 65536 tokens)
//   codebooks: [8,1024,64] fp32
//   out:       [B,N,64]  fp32  = sum over stages of selected codewords
//
// Per wave32: TWO 16-token row tiles share each 16-codeword B tile
//   -> 32 WMMAs per 16 B-loads per chunk, two independent accumulator
//      chains to hide V_WMMA_F32_16X16X4_F32 latency.
// dist(k) = ||c_k||^2 - 2 <r, c_k>  (||r||^2 constant under argmin)
// argmin kept as running (score,idx) per lane, reduced with shfl_xor,
// ties resolved toward the lower codebook index (jnp.argmin semantics).

#define QN    8
#define KCB   1024
#define DIMV  64

typedef __attribute__((ext_vector_type(2))) float v2f;
typedef __attribute__((ext_vector_type(8))) float v8f;

__global__ __launch_bounds__(256)
void ResidualVQ_csq_kernel(const float* __restrict__ cbs,
                           float* __restrict__ csq, int total) {
    int i = blockIdx.x * blockDim.x + threadIdx.x;
    if (i >= total) return;
    const float* r = cbs + (size_t)i * DIMV;
    float s = 0.0f;
#pragma unroll
    for (int k = 0; k < DIMV; ++k) s = __builtin_fmaf(r[k], r[k], s);
    csq[i] = s;
}

__global__ __launch_bounds__(128)
void ResidualVQ_main_kernel(const float* __restrict__ x,
                            const float* __restrict__ cbs,
                            const float* __restrict__ csq,
                            float* __restrict__ out, int tokens) {
    const int lane = threadIdx.x & 31;
    const int wave = threadIdx.x >> 5;
    const int m    = lane & 15;        // row within a 16-token tile (A: M = lane&15)
    const int hi   = lane >> 4;        // upper half carries K pair {2,3} of each step
    const int tileBase = (blockIdx.x * 4 + wave) * 32;   // 32 tokens per wave
    const int token0 = tileBase + m;
    const int token1 = tileBase + 16 + m;

    // Residuals in A-fragment layout: a[j] holds dims {4j+2hi, 4j+2hi+1} of row m.
    const float* xrow0 = x + (size_t)token0 * DIMV;
    const float* xrow1 = x + (size_t)token1 * DIMV;
    v2f a0[16], a1[16], acc0[16], acc1[16];
#pragma unroll
    for (int j = 0; j < 16; ++j) {
        a0[j]   = *(const v2f*)(xrow0 + 4 * j + 2 * hi);
        a1[j]   = *(const v2f*)(xrow1 + 4 * j + 2 * hi);
        acc0[j] = (v2f){0.0f, 0.0f};
        acc1[j] = (v2f){0.0f, 0.0f};
    }

    for (int q = 0; q < QN; ++q) {
        const float* cb = cbs + (size_t)q * KCB * DIMV;
        const float* cs = csq + q * KCB;

        float bestS0[8], bestS1[8];
        int   bestI0[8], bestI1[8];
#pragma unroll
        for (int r = 0; r < 8; ++r) {
            bestS0[r] = __builtin_inff(); bestI0[r] = 0;
            bestS1[r] = __builtin_inff(); bestI1[r] = 0;
        }

        for (int chunk = 0; chunk < KCB / 16; ++chunk) {
            const int n = chunk * 16 + m;                 // this lane's B column
            const float* crow = cb + (size_t)n * DIMV + 2 * hi;

            v2f b[16];
#pragma unroll
            for (int j = 0; j < 16; ++j) b[j] = *(const v2f*)(crow + 4 * j);

            v8f c0 = {};
            v8f c1 = {};
#pragma unroll
            for (int j = 0; j < 16; ++j) {
                c0 = __builtin_amdgcn_wmma_f32_16x16x4_f32(
                         false, a0[j], false, b[j], (short)0, c0, false, false);
                c1 = __builtin_amdgcn_wmma_f32_16x16x4_f32(
                         false, a1[j], false, b[j], (short)0, c1, false, false);
            }

            if (chunk + 1 < KCB / 16)
                __builtin_prefetch(crow + 16 * DIMV, 0, 0); // next chunk's row

            const float cnorm = cs[n];
#pragma unroll
            for (int r = 0; r < 8; ++r) {
                float s0 = __builtin_fmaf(-2.0f, c0[r], cnorm);
                float s1 = __builtin_fmaf(-2.0f, c1[r], cnorm);
                if (s0 < bestS0[r]) { bestS0[r] = s0; bestI0[r] = n; }
                if (s1 < bestS1[r]) { bestS1[r] = s1; bestI1[r] = n; }
            }
        }

        // Argmin across the 16 lanes of each half (width=16 keeps halves apart).
#pragma unroll
        for (int r = 0; r < 8; ++r) {
            float s0 = bestS0[r]; int i0 = bestI0[r];
            float s1 = bestS1[r]; int i1 = bestI1[r];
#pragma unroll
            for (int off = 8; off >= 1; off >>= 1) {
                float so0 = __shfl_xor(s0, off, 16);
                int   io0 = __shfl_xor(i0, off, 16);
                float so1 = __shfl_xor(s1, off, 16);
                int   io1 = __shfl_xor(i1, off, 16);
                if (so0 < s0 || (so0 == s0 && io0 < i0)) { s0 = so0; i0 = io0; }
                if (so1 < s1 || (so1 == s1 && io1 < i1)) { s1 = so1; i1 = io1; }
            }
            bestI0[r] = i0;
            bestI1[r] = i1;
        }

        // Broadcast winners: lower half holds rows 0..7, upper half rows 8..15.
        int myIdx0 = 0, myIdx1 = 0;
#pragma unroll
        for (int r = 0; r < 8; ++r) {
            int lo0 = __shfl(bestI0[r], 0, 32);    // tile0 row r
            int hi0 = __shfl(bestI0[r], 16, 32);   // tile0 row r + 8
            int lo1 = __shfl(bestI1[r], 0, 32);    // tile1 row r
            int hi1 = __shfl(bestI1[r], 16, 32);   // tile1 row r + 8
            if (m == r)     { myIdx0 = lo0; myIdx1 = lo1; }
            if (m == r + 8) { myIdx0 = hi0; myIdx1 = hi1; }
        }

        // Gather selected codewords (L2-resident), update residuals + outputs.
        const float* sel0 = cb + (size_t)myIdx0 * DIMV + 2 * hi;
        const float* sel1 = cb + (size_t)myIdx1 * DIMV + 2 * hi;
#pragma unroll
        for (int j = 0; j < 16; ++j) {
            v2f cv0 = *(const v2f*)(sel0 + 4 * j);
            v2f cv1 = *(const v2f*)(sel1 + 4 * j);
            a0[j]   -= cv0;
            a1[j]   -= cv1;
            acc0[j] += cv0;
            acc1[j] += cv1;
        }
    }

    float* orow0 = out + (size_t)token0 * DIMV;
    float* orow1 = out + (size_t)token1 * DIMV;
#pragma unroll
    for (int j = 0; j < 16; ++j) {
        *(v2f*)(orow0 + 4 * j + 2 * hi) = acc0[j];
        *(v2f*)(orow1 + 4 * j + 2 * hi) = acc1[j];
    }
}

extern "C" void kernel_launch(void* const* d_in, const int* in_sizes, int n_in,
                              void* d_out, int out_size, void* d_ws, size_t ws_size,
                              hipStream_t stream) {
    const float* x   = (const float*)d_in[0];
    const float* cbs = (const float*)d_in[1];
    float*       out = (float*)d_out;
    float*       csq = (float*)d_ws;              // Q*K floats = 32 KB scratch

    const int tokens  = in_sizes[0] / DIMV;       // 65536
    const int totalCB = in_sizes[1] / DIMV;       // 8192 codebook rows

    ResidualVQ_csq_kernel<<<(totalCB + 255) / 256, 256, 0, stream>>>(cbs, csq, totalCB);
    ResidualVQ_main_kernel<<<tokens / 128, 128, 0, stream>>>(x, cbs, csq, out, tokens);
}